// TinyOpsBlock_89094801588364
// MI455X (gfx1250) — compile-verified
//
#include <hip/hip_runtime.h>

// Fused 30x {pointwise, LayerNorm(C=256), Linear 256->256} for MI455X (gfx1250).
// One block persistently owns a 16-row tile. Activations live in LDS; each
// wave's W slice lives in registers (loaded once); GEMM uses
// V_WMMA_F32_16X16X4_F32 with split-K across wave pairs:
//   wave w: khalf = w>>3 (K 0..127 or 128..255), col tiles {2*(w&7), 2*(w&7)+1}
// Each A fragment load now feeds 2 WMMAs (2 col tiles), halving LDS A traffic.

typedef __attribute__((ext_vector_type(2))) float v2f;
typedef __attribute__((ext_vector_type(8))) float v8f;

#define CDIM   256
#define PITCH  260              // 256 + 4 pad -> conflict-free LDS banking
#define ROWS   16               // rows per block (= one WMMA M tile)
#define NITER  30
#define KT     32               // k-tiles (of 4) per wave = half of K
#define RPITCH 20               // row pitch inside a partial tile (bank-safe)
#define RTSZ   (16 * RPITCH)    // floats per 16x16 partial tile

__global__ __launch_bounds__(512, 1) void fused_tinyops_kernel(
    const float* __restrict__ x, const float* __restrict__ ln_w,
    const float* __restrict__ ln_b, const float* __restrict__ W,
    const float* __restrict__ bias, float* __restrict__ out)
{
    __shared__ float hs[ROWS * PITCH];
    __shared__ float red[16 * RTSZ];      // K-upper partial tiles (20 KB)

    const int tid   = threadIdx.x;
    const int lane  = tid & 31;
    const int wave  = tid >> 5;           // 0..15
    const int l16   = lane & 15;
    const int half  = (lane < 16) ? 0 : 2;  // K sub-offset inside a 4-wide k-tile
    const int khalf = wave >> 3;          // 0: K 0..127, 1: K 128..255
    const int cpair = wave & 7;           // owns col tiles 2*cpair, 2*cpair+1
    const int c0    = 2 * cpair;
    const int c1    = c0 + 1;

    const long rowBase = (long)blockIdx.x * ROWS;
    const float* xg = x + rowBase * CDIM;

    // ---- load x tile into LDS (coalesced) ----
    #pragma unroll
    for (int j = 0; j < (ROWS * CDIM) / 512; ++j) {
        int idx = j * 512 + tid;
        hs[(idx >> 8) * PITCH + (idx & (CDIM - 1))] = xg[idx];
    }

    // ---- preload this wave's W slices as B fragments (registers, once) ----
    // B[k][n] = W[16*ct + n][k]; lane covers n = lane&15,
    // k in {4t+half, 4t+half+1} within this wave's K half.
    v2f B0[KT], B1[KT];
    {
        const float* w0 = W + (16 * c0 + l16) * CDIM + 128 * khalf + half;
        const float* w1 = W + (16 * c1 + l16) * CDIM + 128 * khalf + half;
        #pragma unroll
        for (int t = 0; t < KT; ++t) {
            B0[t].x = w0[4 * t + 0];  B0[t].y = w0[4 * t + 1];
            B1[t].x = w1[4 * t + 0];  B1[t].y = w1[4 * t + 1];
        }
    }
    const float bias0 = bias[16 * c0 + l16];
    const float bias1 = bias[16 * c1 + l16];

    // ---- preload LN params for this lane's columns (c = lane + 32*j) ----
    float lw[8], lb[8];
    #pragma unroll
    for (int j = 0; j < 8; ++j) {
        lw[j] = ln_w[lane + 32 * j];
        lb[j] = ln_b[lane + 32 * j];
    }

    __syncthreads();

    for (int it = 0; it < NITER; ++it) {
        // ---- pointwise + LayerNorm: wave r owns row r (wave-private, in place) ----
        float v[8];
        float* hrow = &hs[wave * PITCH];
        #pragma unroll
        for (int j = 0; j < 8; ++j) {
            float h = hrow[lane + 32 * j];
            h = h + 0.01f * tanhf(h);
            h = h * (1.0f / (1.0f + __expf(-h)));
            v[j] = h;
        }
        float s = ((v[0] + v[1]) + (v[2] + v[3])) + ((v[4] + v[5]) + (v[6] + v[7]));
        #pragma unroll
        for (int off = 16; off > 0; off >>= 1) s += __shfl_xor(s, off, 32);
        const float mu = s * (1.0f / CDIM);
        float q = 0.0f;
        #pragma unroll
        for (int j = 0; j < 8; ++j) { float d = v[j] - mu; q += d * d; }
        #pragma unroll
        for (int off = 16; off > 0; off >>= 1) q += __shfl_xor(q, off, 32);
        const float rstd = rsqrtf(q * (1.0f / CDIM) + 1e-5f);
        #pragma unroll
        for (int j = 0; j < 8; ++j)
            hrow[lane + 32 * j] = (v[j] - mu) * rstd * lw[j] + lb[j];

        __syncthreads();

        // ---- GEMM: two 16x16 col tiles over this wave's K half ----
        v8f acc0 = {0.f, 0.f, 0.f, 0.f, 0.f, 0.f, 0.f, 0.f};
        v8f acc1 = {0.f, 0.f, 0.f, 0.f, 0.f, 0.f, 0.f, 0.f};
        const float* arow = &hs[l16 * PITCH + 128 * khalf + half];
        #pragma unroll
        for (int t = 0; t < KT; ++t) {
            v2f a;
            a.x = arow[4 * t + 0];
            a.y = arow[4 * t + 1];
            acc0 = __builtin_amdgcn_wmma_f32_16x16x4_f32(
                false, a, false, B0[t], (short)0, acc0, false, false);
            acc1 = __builtin_amdgcn_wmma_f32_16x16x4_f32(
                false, a, false, B1[t], (short)0, acc1, false, false);
        }

        // ---- K-upper waves publish partials (before the post-GEMM barrier) ----
        if (khalf == 1) {
            float* r0 = &red[(2 * cpair + 0) * RTSZ];
            float* r1 = &red[(2 * cpair + 1) * RTSZ];
            #pragma unroll
            for (int vi = 0; vi < 8; ++vi) {
                int r = vi + ((lane < 16) ? 0 : 8);
                r0[r * RPITCH + l16] = acc0[vi];
                r1[r * RPITCH + l16] = acc1[vi];
            }
        }
        __syncthreads();   // all A reads done + partials visible

        // ---- K-lower waves reduce, add bias, write next activation ----
        if (khalf == 0) {
            const float* r0 = &red[(2 * cpair + 0) * RTSZ];
            const float* r1 = &red[(2 * cpair + 1) * RTSZ];
            #pragma unroll
            for (int vi = 0; vi < 8; ++vi) {
                int r = vi + ((lane < 16) ? 0 : 8);
                hs[r * PITCH + 16 * c0 + l16] =
                    acc0[vi] + r0[r * RPITCH + l16] + bias0;
                hs[r * PITCH + 16 * c1 + l16] =
                    acc1[vi] + r1[r * RPITCH + l16] + bias1;
            }
        }
        __syncthreads();   // new activations visible for next iteration
    }

    // ---- write final activations to d_out (coalesced) ----
    float* og = out + rowBase * CDIM;
    #pragma unroll
    for (int j = 0; j < (ROWS * CDIM) / 512; ++j) {
        int idx = j * 512 + tid;
        og[idx] = hs[(idx >> 8) * PITCH + (idx & (CDIM - 1))];
    }
}

extern "C" void kernel_launch(void* const* d_in, const int* in_sizes, int n_in,
                              void* d_out, int out_size, void* d_ws, size_t ws_size,
                              hipStream_t stream) {
    (void)n_in; (void)d_ws; (void)ws_size; (void)out_size;
    const float* x    = (const float*)d_in[0];
    const float* ln_w = (const float*)d_in[1];
    const float* ln_b = (const float*)d_in[2];
    const float* W    = (const float*)d_in[3];
    const float* b    = (const float*)d_in[4];
    float* out        = (float*)d_out;

    const int total_rows = in_sizes[0] / CDIM;     // 8*2048 = 16384
    dim3 grid(total_rows / ROWS);                  // 1024 blocks
    dim3 block(512);                               // 16 wave32 waves
    fused_tinyops_kernel<<<grid, block, 0, stream>>>(x, ln_w, ln_b, W, b, out);
}